// Model_19739669693008
// MI455X (gfx1250) — compile-verified
//
#include <hip/hip_runtime.h>
#include <math.h>

// ---------------------------------------------------------------------------
// Fused  Q·Qᵀ  +  row-softmax  +  diagonal-zero  +  top-k(20)  for MI455X.
//
// softmax is monotone per row => top-k of probs == top-k of sim (diag excluded).
// Per row we need: rowMax (incl diag), sumExp (incl diag), top-20 sim (excl
// diag).  score_j = exp(v_j - rowMax) / sumExp.
//
// GEMM runs on V_WMMA_F32_16X16X4_F32 (fp32 precision needed for stable
// ordering of the top-k).  Q (16 MB) is L2-resident on a 192 MB L2, so the
// kernel is balanced between the fp32 WMMA pipe and L2 bandwidth, not HBM.
// ---------------------------------------------------------------------------

typedef __attribute__((ext_vector_type(2))) float v2f;
typedef __attribute__((ext_vector_type(8))) float v8f;

#define NQ        8192
#define DIM       512
#define K_TOP     20
#define ROWS_WG   16          // rows (M) per workgroup
#define WAVES     8           // 256 threads / wave32
#define THREADS   256
#define COL_TILES (NQ / 16)           // 512
#define TILES_PW  (COL_TILES / WAVES) // 64 column tiles per wave
#define APAD      (DIM + 4)           // LDS pad: stride 516 dwords -> 4-bank lane stride
#define TPAD      20                  // tile row pad: 16B-aligned, conflict-free b128 reads

__global__ __launch_bounds__(THREADS)
void sim_softmax_topk_kernel(const float* __restrict__ Q, float* __restrict__ out) {
    __shared__ float As[ROWS_WG][APAD];                 // 33.0 KB  A panel (rows of Q)
    __shared__ float tileS[WAVES][16][TPAD];            // 10.2 KB  per-wave C-tile staging
    __shared__ float candV[WAVES][ROWS_WG][K_TOP];      // 10.2 KB  top-k candidate values
    __shared__ int   candI[WAVES][ROWS_WG][K_TOP];      // 10.2 KB  top-k candidate cols
    __shared__ float rowM[WAVES][ROWS_WG];              //  per-wave running row max
    __shared__ float rowS[WAVES][ROWS_WG];              //  per-wave running sum-exp

    const int tid  = threadIdx.x;
    const int wave = tid >> 5;          // wave32
    const int lane = tid & 31;
    const int h    = lane >> 4;         // lane half (K split per ISA f32 frag layout)
    const int lm   = lane & 15;         // M (A frag) / N (B frag) index
    const int rowBase = blockIdx.x * ROWS_WG;

    // ---- stage A panel (16 x 512 f32) into LDS, coalesced float4 ----
    for (int i = tid; i < ROWS_WG * (DIM / 4); i += THREADS) {
        const int r  = i / (DIM / 4);
        const int c4 = i % (DIM / 4);
        const float4 v = ((const float4*)(Q + (size_t)(rowBase + r) * DIM))[c4];
        As[r][c4 * 4 + 0] = v.x;  As[r][c4 * 4 + 1] = v.y;
        As[r][c4 * 4 + 2] = v.z;  As[r][c4 * 4 + 3] = v.w;
    }
    for (int i = tid; i < WAVES * ROWS_WG * K_TOP; i += THREADS) {
        ((float*)candV)[i] = -INFINITY;
        ((int*)candI)[i]   = -1;
    }
    __syncthreads();

    // per-lane persistent stats: lane r (<16) of each wave owns row r
    float runM = -INFINITY;   // running row max (online softmax)
    float runS = 0.0f;        // running sum of exp(v - runM)
    float curMin = -INFINITY; // current 20th-largest candidate

    const float* __restrict__ Ap = &As[lm][2 * h];   // A frag base: As[lm][k0+2h+v]

    for (int t = 0; t < TILES_PW; ++t) {
        const int colBase = (wave * TILES_PW + t) * 16;
        // B frag: b[v] = B[k0+2h+v][lm] = Q[colBase+lm][k0+2h+v]  (L2-resident)
        const float* __restrict__ Bp = Q + (size_t)(colBase + lm) * DIM + 2 * h;

        v8f c = {};
        #pragma unroll 16
        for (int k0 = 0; k0 < DIM; k0 += 4) {
            v2f a, b;
            a.x = Ap[k0];  a.y = Ap[k0 + 1];     // ds_load_b64
            b.x = Bp[k0];  b.y = Bp[k0 + 1];     // global_load_b64 (L2 hit)
            // D = A(16x4) * B(4x16) + C   -- v_wmma_f32_16x16x4_f32
            c = __builtin_amdgcn_wmma_f32_16x16x4_f32(
                    false, a, false, b, (short)0, c, false, false);
        }

        // spill tile: C/D layout -> row M = vg + 8*h, col N = lm
        #pragma unroll
        for (int vg = 0; vg < 8; ++vg)
            tileS[wave][vg + 8 * h][lm] = c[vg];
        // wave-internal cross-lane LDS visibility (no workgroup barrier needed:
        // staging pad is private to this wave)
        asm volatile("s_wait_dscnt 0" ::: "memory");

        // 16 lanes: per-row softmax partials + top-k candidate maintenance.
        // Whole row pulled with 4x ds_load_b128 up front (one wait), tile max
        // computed first so the 16 exps are independent on the trans pipe.
        if (lane < ROWS_WG) {
            const int r    = lane;
            const int grow = rowBase + r;
            const float4 p0 = *(const float4*)&tileS[wave][r][0];
            const float4 p1 = *(const float4*)&tileS[wave][r][4];
            const float4 p2 = *(const float4*)&tileS[wave][r][8];
            const float4 p3 = *(const float4*)&tileS[wave][r][12];
            const float v[16] = { p0.x, p0.y, p0.z, p0.w,  p1.x, p1.y, p1.z, p1.w,
                                  p2.x, p2.y, p2.z, p2.w,  p3.x, p3.y, p3.z, p3.w };
            float tmax = v[0];
            #pragma unroll
            for (int n = 1; n < 16; ++n) tmax = fmaxf(tmax, v[n]);
            float tsum = 0.0f;
            #pragma unroll
            for (int n = 0; n < 16; ++n) tsum += __expf(v[n] - tmax);
            // single online-softmax merge per tile (diagonal INCLUDED: the
            // reference zeroes probs only after softmax, so diag stays in
            // max & denominator)
            if (tmax > runM) { runS = runS * __expf(runM - tmax) + tsum; runM = tmax; }
            else             { runS += tsum * __expf(tmax - runM); }
            // top-k inserts (diagonal EXCLUDED); rare: ~100 per row-wave total
            #pragma unroll
            for (int n = 0; n < 16; ++n) {
                const int col = colBase + n;
                if (col != grow && v[n] > curMin) {
                    int slot = 0; float mn = candV[wave][r][0];
                    #pragma unroll
                    for (int j = 1; j < K_TOP; ++j) {
                        const float cv = candV[wave][r][j];
                        if (cv < mn) { mn = cv; slot = j; }
                    }
                    candV[wave][r][slot] = v[n];
                    candI[wave][r][slot] = col;
                    mn = candV[wave][r][0];
                    #pragma unroll
                    for (int j = 1; j < K_TOP; ++j) mn = fminf(mn, candV[wave][r][j]);
                    curMin = mn;
                }
            }
        }
    }

    if (lane < ROWS_WG) { rowM[wave][lane] = runM; rowS[wave][lane] = runS; }
    __syncthreads();

    // ---- merge the 8 waves' per-row stats + candidates, emit outputs ----
    if (tid < ROWS_WG) {
        const int r    = tid;
        const int grow = rowBase + r;
        float M = -INFINITY;
        #pragma unroll
        for (int w = 0; w < WAVES; ++w) M = fmaxf(M, rowM[w][r]);
        float S = 0.0f;
        #pragma unroll
        for (int w = 0; w < WAVES; ++w) S += rowS[w][r] * __expf(rowM[w][r] - M);
        const float invS = 1.0f / S;

        float* outScore = out;
        float* outInd   = out + (size_t)NQ * K_TOP;
        float* outMask  = out + (size_t)2 * NQ * K_TOP;
        float* outCnt   = out + (size_t)3 * NQ * K_TOP;

        int count = 0;
        for (int j = 0; j < K_TOP; ++j) {          // selection => descending order
            float best = -INFINITY; int bw = 0, bs = 0;
            for (int w = 0; w < WAVES; ++w)
                #pragma unroll
                for (int s = 0; s < K_TOP; ++s) {
                    const float v = candV[w][r][s];
                    if (v > best) { best = v; bw = w; bs = s; }
                }
            const int col = candI[bw][r][bs];
            candV[bw][r][bs] = -INFINITY;
            const float score = __expf(best - M) * invS;
            const bool  m     = score > 0.0f;      // THRESHOLD = 0.0
            outScore[(size_t)grow * K_TOP + j] = score;
            outInd  [(size_t)grow * K_TOP + j] = (float)col;
            outMask [(size_t)grow * K_TOP + j] = m ? 1.0f : 0.0f;
            count += m ? 1 : 0;
        }
        outCnt[grow] = (float)count;
    }
}

extern "C" void kernel_launch(void* const* d_in, const int* in_sizes, int n_in,
                              void* d_out, int out_size, void* d_ws, size_t ws_size,
                              hipStream_t stream) {
    (void)in_sizes; (void)n_in; (void)d_ws; (void)ws_size; (void)out_size;
    const float* Q = (const float*)d_in[0];   // queries [8192, 512] f32
    // d_in[1] is k (== 20, compile-time K_TOP)
    float* out = (float*)d_out;               // scores | inds | mask | counts (flat)
    dim3 grid(NQ / ROWS_WG);                  // 512 workgroups
    dim3 block(THREADS);                      // 8 wave32s
    sim_softmax_topk_kernel<<<grid, block, 0, stream>>>(Q, out);
}